// CustomInteractionBlock_2293512536751
// MI455X (gfx1250) — compile-verified
//
#include <hip/hip_runtime.h>
#include <cstddef>

// ---------------- problem constants ----------------
#define NN      25000
#define EE      200000
#define MULM    16
#define HIDDEN  64
#define WNUMEL  1024            // 4*16*16
#define NTILES  (EE / 16)       // 12500 (E divisible by 16)

#define PATH_ALPHA 0.17677669529663689f   // 1/sqrt(32)
#define INV_SQRT3  0.57735026918962580f
#define INV_SQRT8  0.35355339059327380f
#define W2_SCALE   0.125f                 // 1/sqrt(64)
#define SC_SCALE   0.25f                  // 1/sqrt(16)
#define CEN_STEP   0.71428571428571430f   // 5/7

#define WAVES_PER_WG 8
#define BLOCK        (WAVES_PER_WG * 32)
#define NUM_WG       250

// ---------------- LDS layout (floats) ----------------
#define W2_F      65536                   // 64x1024 pre-swizzled B blocks
#define W1_F      512
#define WAVE_F    1408                    // per-wave scratch
#define XJ_OFF    0                       // 16 edges x 64 ch
#define A1_OFF    1024                    // 16x16 a1 table
#define SH_OFF    1280                    // 4 x 16 sh values
#define LEN_OFF   1344                    // 16 lengths
#define SRC_OFF   1360                    // 16 ints
#define DST_OFF   1376                    // 16 ints
#define SMEM_F    (W2_F + W1_F + WAVES_PER_WG * WAVE_F)   // 77312 floats = 302 KB

typedef float v2f __attribute__((ext_vector_type(2)));
typedef float v8f __attribute__((ext_vector_type(8)));

// Wave-level LDS sync: order + wait for this wave's DS ops (per-wave private
// scratch regions, so no cross-wave barrier needed).
__device__ __forceinline__ void wave_sync() {
    __builtin_amdgcn_wave_barrier();
    asm volatile("s_wait_dscnt 0" ::: "memory");
    __builtin_amdgcn_wave_barrier();
}

__device__ __forceinline__ float silu(float z) {
    return z / (1.0f + __expf(-z));
}

// ---------------- kernel 1: out = self-connection (init) ----------------
__global__ __launch_bounds__(256) void sc_init_kernel(
    const float* __restrict__ x, const float* __restrict__ L0,
    const float* __restrict__ L1, float* __restrict__ out)
{
    int idx = blockIdx.x * blockDim.x + threadIdx.x;
    if (idx >= NN * 64) return;
    int n = idx >> 6, col = idx & 63;
    const float* xr = x + (size_t)n * 64;
    float acc = 0.0f;
    if (col < 16) {
        int v = col;
        #pragma unroll
        for (int u = 0; u < 16; ++u) acc += xr[u] * L0[u * 16 + v];
    } else {
        int c = col - 16;
        int v = c / 3, k = c - 3 * v;
        #pragma unroll
        for (int u = 0; u < 16; ++u) acc += xr[16 + 3 * u + k] * L1[u * 16 + v];
    }
    out[idx] = acc * SC_SCALE;
}

// ---------------- kernel 2: fused edge interaction ----------------
__global__ __launch_bounds__(BLOCK, 1) void edge_kernel(
    const float* __restrict__ xg,        const float* __restrict__ edge_attr,
    const float* __restrict__ edge_len,  const int*  __restrict__ edge_src,
    const int*  __restrict__ edge_dst,   const float* __restrict__ W1g,
    const float* __restrict__ W2g,       float* __restrict__ out)
{
    extern __shared__ float smem[];
    float* w2s = smem;                 // [ (nt*16+kk)*64 + 2*lane + {0,1} ]
    float* w1s = smem + W2_F;          // [r*64 + c]

    const int tid  = threadIdx.x;
    const int lane = tid & 31;
    const int wv   = tid >> 5;
    const int hi   = lane >> 4;        // lane half
    const int j    = lane & 15;        // column / edge-slot index

    float* wbase = smem + W2_F + W1_F + wv * WAVE_F;
    float* xjs  = wbase + XJ_OFF;
    float* a1s  = wbase + A1_OFF;
    float* shs  = wbase + SH_OFF;
    float* lens = wbase + LEN_OFF;
    int*   srcs = (int*)(wbase + SRC_OFF);
    int*   dsts = (int*)(wbase + DST_OFF);

    // ---- stage W2 (scaled, pre-swizzled into per-WMMA B-operand blocks) ----
    // block b = nt*16+kk holds B[k=0..3][n=0..15]; lane l keeps rows 2*hi,2*hi+1
    for (int idx = tid; idx < W2_F; idx += BLOCK) {
        int b   = idx >> 6;
        int pos = idx & 63;
        int l   = pos >> 1, w = pos & 1;
        int lh  = l >> 4,  lj = l & 15;
        int kk  = b & 15,  nt = b >> 4;
        int c   = 4 * kk + 2 * lh + w;      // hidden channel (K row)
        int n   = 16 * nt + lj;             // w column
        w2s[idx] = W2g[c * WNUMEL + n] * W2_SCALE;
    }
    for (int idx = tid; idx < W1_F; idx += BLOCK) w1s[idx] = W1g[idx];
    __syncthreads();

    // ---- per-wave independent edge-tile loop ----
    for (int tile = blockIdx.x * WAVES_PER_WG + wv; tile < NTILES;
         tile += gridDim.x * WAVES_PER_WG) {
        const int e0 = tile * 16;

        wave_sync();  // previous tile's LDS reads done before restaging

        // stage per-edge scalars (lanes 0..15)
        if (lane < 16) {
            int e = e0 + lane;
            srcs[lane] = edge_src[e];
            dsts[lane] = edge_dst[e];
            lens[lane] = edge_len[e];
            #pragma unroll
            for (int c = 0; c < 4; ++c) shs[c * 16 + lane] = edge_attr[e * 4 + c];
        }
        wave_sync();

        // coalesced gather of xj = x[src] (16 edges x 64 floats)
        #pragma unroll 4
        for (int e = 0; e < 16; ++e) {
            int s = srcs[e];
            const float* xr = xg + (size_t)s * 64;
            xjs[e * 64 + lane]      = xr[lane];
            xjs[e * 64 + 32 + lane] = xr[32 + lane];
        }
        wave_sync();

        // a1[e][u] = inv_sqrt3 * sum_k xj1[e,u,k] * sh1[e,k]
        {
            int e = j;
            float s1 = shs[16 + e], s2 = shs[32 + e], s3 = shs[48 + e];
            #pragma unroll
            for (int i = 0; i < 8; ++i) {
                int u = 8 * hi + i;
                int r = e * 64 + 16 + 3 * u;
                a1s[e * 16 + u] =
                    INV_SQRT3 * (xjs[r] * s1 + xjs[r + 1] * s2 + xjs[r + 2] * s3);
            }
        }
        wave_sync();

        // ---- h = silu(radial @ W1 / sqrt(8)), in WMMA A layout ----
        // A[kk]: lane holds h[e=j, c=4*kk+2*hi+{0,1}]
        float len = lens[j];
        float rad[8];
        #pragma unroll
        for (int r = 0; r < 8; ++r) {
            float d = len - (float)r * CEN_STEP;
            rad[r] = __expf(-0.5f * d * d);
        }
        v2f A[16];
        #pragma unroll
        for (int kk = 0; kk < 16; ++kk) {
            int c0 = 4 * kk + 2 * hi;
            float z0 = 0.0f, z1 = 0.0f;
            #pragma unroll
            for (int r = 0; r < 8; ++r) {
                z0 += rad[r] * w1s[r * 64 + c0];
                z1 += rad[r] * w1s[r * 64 + c0 + 1];
            }
            A[kk].x = silu(z0 * INV_SQRT8);
            A[kk].y = silu(z1 * INV_SQRT8);
        }

        // ---- accumulators (C/D layout: VGPR i -> edge 8*hi+i, col j) ----
        v8f s0a = {0,0,0,0,0,0,0,0};   // path0: sum_u xj0*w0
        v8f s1a = {0,0,0,0,0,0,0,0};   // path1: sum_u a1*w1
        v8f t2a = {0,0,0,0,0,0,0,0};   // path2: sum_u xj0*w2
        v8f t30 = {0,0,0,0,0,0,0,0};   // path3 k=0..2: sum_u xj1k*w3
        v8f t31 = {0,0,0,0,0,0,0,0};
        v8f t32 = {0,0,0,0,0,0,0,0};

        const int ebase = 8 * hi;

        // ---- hot loop: 64 output tiles, 16 chained f32 WMMAs each ----
        for (int nt = 0; nt < 64; ++nt) {
            const float* bb = w2s + nt * 1024 + 2 * lane;
            v8f d = {0,0,0,0,0,0,0,0};
            #pragma unroll
            for (int kk = 0; kk < 16; ++kk) {
                v2f b = *(const v2f*)(bb + kk * 64);
                d = __builtin_amdgcn_wmma_f32_16x16x4_f32(
                        false, A[kk], false, b, (short)0, d, false, false);
            }
            int p = nt >> 4, u = nt & 15;
            if (p == 0) {
                #pragma unroll
                for (int i = 0; i < 8; ++i)
                    s0a[i] += xjs[(ebase + i) * 64 + u] * d[i];
            } else if (p == 1) {
                #pragma unroll
                for (int i = 0; i < 8; ++i)
                    s1a[i] += a1s[(ebase + i) * 16 + u] * d[i];
            } else if (p == 2) {
                #pragma unroll
                for (int i = 0; i < 8; ++i)
                    t2a[i] += xjs[(ebase + i) * 64 + u] * d[i];
            } else {
                #pragma unroll
                for (int i = 0; i < 8; ++i) {
                    int r = (ebase + i) * 64 + 16 + 3 * u;
                    float dv = d[i];
                    t30[i] += xjs[r] * dv;
                    t31[i] += xjs[r + 1] * dv;
                    t32[i] += xjs[r + 2] * dv;
                }
            }
        }

        // ---- finalize + scatter-add by edge_dst ----
        #pragma unroll
        for (int i = 0; i < 8; ++i) {
            int e = ebase + i;
            float sh0 = shs[e];
            float s11 = shs[16 + e], s12 = shs[32 + e], s13 = shs[48 + e];
            float* orow = out + (size_t)dsts[e] * 64;
            atomicAdd(orow + j, PATH_ALPHA * (sh0 * s0a[i] + s1a[i]));
            float t2v = t2a[i];
            atomicAdd(orow + 16 + 3 * j + 0, PATH_ALPHA * (s11 * t2v + sh0 * t30[i]));
            atomicAdd(orow + 16 + 3 * j + 1, PATH_ALPHA * (s12 * t2v + sh0 * t31[i]));
            atomicAdd(orow + 16 + 3 * j + 2, PATH_ALPHA * (s13 * t2v + sh0 * t32[i]));
        }
    }
}

// ---------------- host launcher ----------------
extern "C" void kernel_launch(void* const* d_in, const int* in_sizes, int n_in,
                              void* d_out, int out_size, void* d_ws, size_t ws_size,
                              hipStream_t stream) {
    const float* x         = (const float*)d_in[0];
    const float* edge_attr = (const float*)d_in[1];
    const float* edge_len  = (const float*)d_in[2];
    const int*   edge_src  = (const int*)d_in[3];
    const int*   edge_dst  = (const int*)d_in[4];
    const float* W1        = (const float*)d_in[5];
    const float* W2        = (const float*)d_in[6];
    const float* L0        = (const float*)d_in[7];
    const float* L1        = (const float*)d_in[8];
    float* out = (float*)d_out;

    // 1) initialize out with the self-connection
    sc_init_kernel<<<(NN * 64 + 255) / 256, 256, 0, stream>>>(x, L0, L1, out);

    // 2) fused edge interaction with atomic scatter on top
    size_t shmem = (size_t)SMEM_F * sizeof(float);   // ~302 KB (< 320 KB/WGP)
    (void)hipFuncSetAttribute((const void*)edge_kernel,
                              hipFuncAttributeMaxDynamicSharedMemorySize,
                              (int)shmem);
    edge_kernel<<<NUM_WG, BLOCK, shmem, stream>>>(
        x, edge_attr, edge_len, edge_src, edge_dst, W1, W2, out);
}